// ScaledDotProductAttention_82489141887459
// MI455X (gfx1250) — compile-verified
//
#include <hip/hip_runtime.h>
#include <cstdint>

typedef __attribute__((ext_vector_type(2))) float v2f;
typedef __attribute__((ext_vector_type(8))) float v8f;

namespace {
constexpr int kB = 2;
constexpr int kS = 2048;     // sequence length (both L and S)
constexpr int kH = 16;       // heads
constexpr int kE = 64;       // head dim
constexpr int kBQ = 16;      // query rows per workgroup
constexpr int kWaves = 4;    // 128 threads, wave32
constexpr int kThreads = kWaves * 32;
constexpr float kScale = 0.125f;              // 1/sqrt(64)
constexpr float kKeepScale = 1.0f / 0.8f;     // 1/(1-p)
constexpr uint32_t kKeepThresh = 13421773u;   // floor(0.8 * 2^24)
}  // namespace

__device__ __forceinline__ uint32_t mix64(uint64_t x) {
  x ^= x >> 33; x *= 0xff51afd7ed558ccdULL;
  x ^= x >> 33; x *= 0xc4ceb9fe1a85ec53ULL;
  x ^= x >> 33;
  return (uint32_t)x;
}

__global__ __launch_bounds__(kThreads)
void sdpa_fwd_f32wmma(const float* __restrict__ q,
                      const float* __restrict__ kmat,
                      const float* __restrict__ vmat,
                      float* __restrict__ ctx,
                      float* __restrict__ attn) {
  extern __shared__ float sS[];  // [kBQ][kS] scores -> probs -> dropped probs

  const int tid = threadIdx.x;
  const int wave = tid >> 5;
  const int lane = tid & 31;
  const int m = lane & 15;       // row/col within 16-tile
  const int khalf = lane >> 4;   // which K-pair half (ISA A/B layout)

  const int nLBlk = kS / kBQ;    // 128
  const int blk = blockIdx.x;    // over B*H*nLBlk
  const int lblk = blk % nLBlk;
  const int bh = blk / nLBlk;
  const int h = bh % kH;
  const int b = bh / kH;
  const int l0 = lblk * kBQ;

  const size_t sStride = (size_t)kH * kE;  // 1024 floats between seq positions
  const float* qb = q    + ((size_t)b * kS) * sStride + (size_t)h * kE;
  const float* kb = kmat + ((size_t)b * kS) * sStride + (size_t)h * kE;
  const float* vb = vmat + ((size_t)b * kS) * sStride + (size_t)h * kE;

  // ---- Load Q block (16x64) as 16 A-fragments of shape 16x4 ----
  // A layout (32-bit, 16x4): lanes 0-15 hold K={0,1}, lanes 16-31 hold K={2,3}
  v2f aq[16];
  {
    const float* qrow = qb + (size_t)(l0 + m) * sStride + 2 * khalf;
#pragma unroll
    for (int c = 0; c < 16; ++c)
      aq[c] = *(const v2f*)(qrow + 4 * c);
  }

  // ---- Phase 1: S = Q @ K^T, tiles of 16 keys, scores kept in LDS ----
  const int tilesPerWave = (kS / 16) / kWaves;  // 32
  for (int t = 0; t < tilesPerWave; ++t) {
    const int n0 = (wave * tilesPerWave + t) * 16;
    const float* krow = kb + (size_t)(n0 + m) * sStride + 2 * khalf;
    v8f acc = {};
#pragma unroll
    for (int c = 0; c < 16; ++c) {
      v2f bk = *(const v2f*)(krow + 4 * c);
      acc = __builtin_amdgcn_wmma_f32_16x16x4_f32(false, aq[c], false, bk,
                                                  (short)0, acc, false, false);
    }
    // C/D layout: VGPR r -> M = r + 8*khalf, N = m
#pragma unroll
    for (int r = 0; r < 8; ++r)
      sS[(size_t)(r + 8 * khalf) * kS + (n0 + m)] = acc[r];
  }
  __syncthreads();

  // ---- Phase 2: per-row softmax + dropout; each wave owns 4 rows ----
  // Lane-contiguous sweeps: conflict-free LDS banks, coalesced 128B HBM stores.
  for (int rr = 0; rr < kBQ / kWaves; ++rr) {
    const int row = wave * (kBQ / kWaves) + rr;
    float* srow = sS + (size_t)row * kS;

    float mx = -3.402823466e+38f;
    for (int n = lane; n < kS; n += 32)
      mx = fmaxf(mx, srow[n]);
#pragma unroll
    for (int off = 16; off > 0; off >>= 1)
      mx = fmaxf(mx, __shfl_xor(mx, off, 32));

    float sum = 0.0f;
    for (int n = lane; n < kS; n += 32) {
      float e = __expf(kScale * (srow[n] - mx));
      srow[n] = e;
      sum += e;
    }
#pragma unroll
    for (int off = 16; off > 0; off >>= 1)
      sum += __shfl_xor(sum, off, 32);

    const float inv = 1.0f / sum;
    const uint64_t rowIdx = ((uint64_t)bh * kS + (uint64_t)(l0 + row)) * (uint64_t)kS;
    float* arow = attn + rowIdx;
    for (int n = lane; n < kS; n += 32) {
      float p = srow[n] * inv;
      uint32_t hsh = mix64(rowIdx + (uint64_t)n);
      float w = ((hsh & 0xFFFFFFu) < kKeepThresh) ? p * kKeepScale : 0.0f;
      srow[n] = w;      // keep post-dropout weight for the context GEMM
      arow[n] = w;      // materialize attention weights (dominant HBM traffic)
    }
  }
  __syncthreads();

  // ---- Phase 3: context = attn(16x2048) @ V(2048x64); wave owns 16 dims ----
  {
    const int d0 = wave * 16;
    const float* vcol = vb + (d0 + m);
    v8f acc = {};
#pragma unroll 4
    for (int c = 0; c < kS / 4; ++c) {
      const int kk = 4 * c + 2 * khalf;
      v2f af = *(const v2f*)(&sS[(size_t)m * kS + kk]);  // A: row m, K pair
      v2f bv;
      const float* vp = vcol + (size_t)kk * sStride;
      bv.x = vp[0];
      bv.y = vp[sStride];
      acc = __builtin_amdgcn_wmma_f32_16x16x4_f32(false, af, false, bv,
                                                  (short)0, acc, false, false);
    }
#pragma unroll
    for (int r = 0; r < 8; ++r) {
      const int M = r + 8 * khalf;  // query row within block
      ctx[(((size_t)b * kS + (size_t)(l0 + M)) * kH + h) * kE + (d0 + m)] = acc[r];
    }
  }
}

extern "C" void kernel_launch(void* const* d_in, const int* /*in_sizes*/, int /*n_in*/,
                              void* d_out, int /*out_size*/, void* /*d_ws*/, size_t /*ws_size*/,
                              hipStream_t stream) {
  const float* q = (const float*)d_in[0];
  const float* k = (const float*)d_in[1];
  const float* v = (const float*)d_in[2];
  float* ctx  = (float*)d_out;                               // [B,L,H,E]
  float* attn = ctx + (size_t)kB * kS * kH * kE;             // [B,H,L,S]

  const int grid = kB * kH * (kS / kBQ);                     // 4096 workgroups
  const size_t ldsBytes = (size_t)kBQ * kS * sizeof(float);  // 128 KB / WG

  hipFuncSetAttribute((const void*)sdpa_fwd_f32wmma,
                      hipFuncAttributeMaxDynamicSharedMemorySize, (int)ldsBytes);
  sdpa_fwd_f32wmma<<<grid, kThreads, ldsBytes, stream>>>(q, k, v, ctx, attn);
}